// CrossNet_24240795418639
// MI455X (gfx1250) — compile-verified
//
#include <hip/hip_runtime.h>

typedef __attribute__((ext_vector_type(2))) float v2f;
typedef __attribute__((ext_vector_type(8))) float v8f;
typedef __attribute__((ext_vector_type(4))) int   v4i;

#define DDIM 4096
#define LAYERS 6
#define NROWS 16384
#define KC 256            // K chunk staged in LDS per iteration
#define PITCH 260         // 256 + 4 floats padding: conflict-free strided b64 A-frag reads

// --- CDNA5 async load-to-LDS path (GLOBAL_LOAD_ASYNC_TO_LDS_B128, ASYNCcnt) ---
#define USE_ASYNC 0
#if defined(__has_builtin)
#if __has_builtin(__builtin_amdgcn_global_load_async_to_lds_b128) && \
    __has_builtin(__builtin_amdgcn_s_wait_asynccnt)
#undef USE_ASYNC
#define USE_ASYNC 1
#endif
#endif

// Builtin signature (from hipcc diagnostic): (v4i AS1*, v4i AS3*, imm int, imm int)
#define AS1_V4I(p) ((__attribute__((address_space(1))) v4i*)(p))
#define AS3_V4I(p) ((__attribute__((address_space(3))) v4i*)(p))

// ---------------------------------------------------------------------------
// Prep kernel (one block): t_j = dot(Bcum_j, w_j); Bcum = sum_j b_j; zero pad
// ---------------------------------------------------------------------------
__global__ __launch_bounds__(256) void crossnet_prep(
    const float* __restrict__ w, const float* __restrict__ b,
    float* __restrict__ tvals, float* __restrict__ bcum, float* __restrict__ zbuf)
{
  __shared__ float red[256];
  const int tid = threadIdx.x;
  float bc[16];
#pragma unroll
  for (int i = 0; i < 16; ++i) bc[i] = 0.f;

  for (int j = 0; j < LAYERS; ++j) {
    float p = 0.f;
#pragma unroll
    for (int i = 0; i < 16; ++i) p += bc[i] * w[j * DDIM + tid + 256 * i];
    red[tid] = p;
    __syncthreads();
    for (int s = 128; s > 0; s >>= 1) {
      if (tid < s) red[tid] += red[tid + s];
      __syncthreads();
    }
    if (tid == 0) tvals[j] = red[0];
    __syncthreads();
#pragma unroll
    for (int i = 0; i < 16; ++i) bc[i] += b[j * DDIM + tid + 256 * i];
  }
#pragma unroll
  for (int i = 0; i < 16; ++i) bcum[tid + 256 * i] = bc[i];
#pragma unroll
  for (int i = 0; i < 16; ++i) zbuf[tid + 256 * i] = 0.f;   // zero B-cols n>=6
}

// ---------------------------------------------------------------------------
// Main kernel: one wave32 per block, 16 rows per wave.
//   Phase 1: a[r][j] = dot(x0_row_r, w_j) via V_WMMA_F32_16X16X4_F32
//   Phase 2: 6-step scalar recurrence -> scale[r] = 1 + S_6
//   Phase 3: out = x0 * scale[r] + Bcum  (x0 re-read hits L2)
// ---------------------------------------------------------------------------
__global__ __launch_bounds__(32) void crossnet_main(
    const float* __restrict__ x0, const float* __restrict__ w,
    const float* __restrict__ tvals, const float* __restrict__ bcum,
    const float* __restrict__ zbuf, float* __restrict__ out)
{
  __shared__ float xs[16 * PITCH];
  __shared__ float cmat[256];
  __shared__ float scale_s[16];

  const int lane = threadIdx.x;
  const size_t row_base = (size_t)blockIdx.x * 16;
  const float* xbase = x0 + row_base * DDIM;

  v8f c = {0.f, 0.f, 0.f, 0.f, 0.f, 0.f, 0.f, 0.f};

  const int n    = lane & 15;            // A: row index; B: column index
  const int koff = (lane >> 4) << 1;     // K sub-pair: {0,1} or {2,3}
  // Lanes n>=6 read a zero page: B columns 6..15 are exactly 0, no mask mul.
  const float* wrow = (n < LAYERS) ? (w + (size_t)n * DDIM) : zbuf;

  for (int kc0 = 0; kc0 < DDIM; kc0 += KC) {
    // ---- stage 16 x KC tile of x0 into LDS ----
#if USE_ASYNC
#pragma unroll 4
    for (int i = lane; i < 16 * (KC / 4); i += 32) {
      const int r  = i >> 6;             // KC/4 = 64 float4 per row
      const int kk = (i & 63) << 2;
      const float* gp = xbase + (size_t)r * DDIM + kc0 + kk;
      float* lp = &xs[r * PITCH + kk];
      __builtin_amdgcn_global_load_async_to_lds_b128(AS1_V4I(gp), AS3_V4I(lp), 0, 0);
    }
    __builtin_amdgcn_s_wait_asynccnt(0);
    __syncthreads();
#else
#pragma unroll 4
    for (int i = lane; i < 16 * (KC / 4); i += 32) {
      const int r  = i >> 6;
      const int kk = (i & 63) << 2;
      const float4 v = *(const float4*)(xbase + (size_t)r * DDIM + kc0 + kk);
      *(float4*)(&xs[r * PITCH + kk]) = v;
    }
    __syncthreads();
#endif

    // ---- KC/4 chained fp32 WMMAs ----
#pragma unroll 4
    for (int k4 = 0; k4 < KC / 4; ++k4) {
      const int kl = (k4 << 2) + koff;
      const v2f a2 = *(const v2f*)(&xs[n * PITCH + kl]);
      const v2f b2 = *(const v2f*)(wrow + kc0 + kl);
      c = __builtin_amdgcn_wmma_f32_16x16x4_f32(
          /*neg_a=*/false, a2, /*neg_b=*/false, b2,
          /*c_mod=*/(short)0, c, /*reuse_a=*/false, /*reuse_b=*/false);
    }
    __syncthreads();
  }

  // ---- dump C (a[r][j] in columns 0..5) to LDS ----
  const int mbase = (lane >> 4) << 3;    // 0 or 8
#pragma unroll
  for (int j = 0; j < 8; ++j) cmat[(mbase + j) * 16 + n] = c[j];
  __syncthreads();

  // ---- per-row recurrence: s_j = (1+S)*a_j + t_j ; S += s_j ----
  if (lane < 16) {
    float S = 0.f;
#pragma unroll
    for (int j = 0; j < LAYERS; ++j) {
      const float s = (1.0f + S) * cmat[lane * 16 + j] + tvals[j];
      S += s;
    }
    scale_s[lane] = 1.0f + S;
  }
  __syncthreads();

  // ---- output pass: out = x0 * scale[row] + Bcum (all lanes same row/iter) ----
#pragma unroll 4
  for (int i = lane; i < 16 * (DDIM / 4); i += 32) {
    const int r  = i >> 10;              // DDIM/4 = 1024 float4 per row
    const int c4 = (i & 1023) << 2;
    const float4 xv = *(const float4*)(xbase + (size_t)r * DDIM + c4);
    const float4 bv = *(const float4*)(bcum + c4);
    const float sc = scale_s[r];
    float4 o;
    o.x = xv.x * sc + bv.x;
    o.y = xv.y * sc + bv.y;
    o.z = xv.z * sc + bv.z;
    o.w = xv.w * sc + bv.w;
    *(float4*)(out + (row_base + r) * DDIM + c4) = o;
  }
}

// ---------------------------------------------------------------------------
extern "C" void kernel_launch(void* const* d_in, const int* in_sizes, int n_in,
                              void* d_out, int out_size, void* d_ws, size_t ws_size,
                              hipStream_t stream) {
  const float* x0 = (const float*)d_in[0];   // [16384, 4096]
  const float* w  = (const float*)d_in[1];   // [6, 4096, 1]
  const float* b  = (const float*)d_in[2];   // [6, 4096]
  float* out = (float*)d_out;                // [16384, 4096]

  float* tvals = (float*)d_ws;               // 6 floats (+pad)
  float* bcum  = (float*)d_ws + 16;          // 4096 floats
  float* zbuf  = (float*)d_ws + 16 + DDIM;   // 4096 floats of zeros

  crossnet_prep<<<1, 256, 0, stream>>>(w, b, tvals, bcum, zbuf);
  crossnet_main<<<NROWS / 16, 32, 0, stream>>>(x0, w, tvals, bcum, zbuf, out);
}